// SphericalEmbedding_21320217657641
// MI455X (gfx1250) — compile-verified
//
#include <hip/hip_runtime.h>
#include <math.h>

typedef float v2f __attribute__((ext_vector_type(2)));
typedef float v8f __attribute__((ext_vector_type(8)));

#define SE_DEPTH 8

// One wave (32 lanes) processes 16 points.
//  - lane n (=lane&15) computes spherical coords of point base+n for all 8 depths
//  - per column-tile t (2 depths): two chained V_WMMA_F32_16X16X4_F32 produce a
//    fully-dense 16x16 f32 output tile (16 points x 16 (depth,dim) columns)
__global__ __launch_bounds__(256) void se_wmma_kernel(
    const float* __restrict__ x,        // [B,3]
    const float* __restrict__ centers,  // [8,3]
    const float* __restrict__ proj,     // [8,3,8]
    float* __restrict__ out,            // [B,8,8]
    int nPoints)
{
  const int lane = threadIdx.x & 31;
  const int wib  = threadIdx.x >> 5;
  const int n    = lane & 15;   // tile column / point-in-tile index
  const int half = lane >> 4;   // 0: lanes 0-15, 1: lanes 16-31

  const long long waveId = (long long)blockIdx.x * (blockDim.x >> 5) + wib;
  const long long base   = waveId * 16;
  if (base >= nPoints) return;          // wave-uniform: EXEC stays all-ones

  const bool fullTile = (base + 16 <= (long long)nPoints);  // wave-uniform

  long long p  = base + n;
  long long pc = (p < nPoints) ? p : (long long)nPoints - 1;

  // streaming hint for the x stream (speculative prefetch; OOB silently dropped)
  __builtin_prefetch(x + pc * 3 + 3 * 16 * 256, 0, 0);

  const float px = x[pc * 3 + 0];
  const float py = x[pc * 3 + 1];
  const float pz = x[pc * 3 + 2];

  float rho[SE_DEPTH], phi[SE_DEPTH], theta[SE_DEPTH];
#pragma unroll
  for (int d = 0; d < SE_DEPTH; ++d) {
    const float dx = px - centers[d * 3 + 0];
    const float dy = py - centers[d * 3 + 1];
    const float dz = pz - centers[d * 3 + 2];
    const float r  = sqrtf(dx * dx + dy * dy + dz * dz);
    rho[d]   = r;
    phi[d]   = atan2f(dy, dx);          // arctan2(y, x)
    theta[d] = acosf(dz / r);           // arccos(z / rho)
  }

  // Column masks as 0.0/1.0 floats: branch-free B-matrix construction.
  const float mlo = (n < 8) ? 1.0f : 0.0f;  // columns 0-7  -> depth d0
  const float mhi = 1.0f - mlo;             // columns 8-15 -> depth d1

#if __has_builtin(__builtin_amdgcn_wmma_f32_16x16x4_f32)
#pragma unroll
  for (int t = 0; t < 4; ++t) {
    const int d0 = 2 * t, d1 = d0 + 1;

    // A tile 16x4 (f32): lanes 0-15 hold K0(v0),K1(v1); lanes 16-31 hold K2,K3.
    // chunk0 K rows: rho_d0, phi_d0, theta_d0, rho_d1
    // chunk1 K rows: phi_d1, theta_d1, 0, 0
    v2f a0, a1;
    a0[0] = half ? theta[d0] : rho[d0];
    a0[1] = half ? rho[d1]   : phi[d0];
    a1[0] = half ? 0.0f      : phi[d1];
    a1[1] = half ? 0.0f      : theta[d1];

    // B tile 4x16 (f32), column = n. Unconditional loads (all indices below are
    // in-bounds of proj[192] for every lane) zeroed by mask multiply -> no
    // exec-mask branches around the loads.
    const int jn = n - 8;                                  // in [-8,7]
    const float r0n = proj[d0 * 24 +  0 + n ] * mlo;       // P[d0,0,n]   | 0
    const float r1n = proj[d0 * 24 +  8 + n ] * mlo;       // P[d0,1,n]   | 0
    const float r2n = proj[d0 * 24 + 16 + n ] * mlo;       // P[d0,2,n]   | 0
    const float r3n = proj[d1 * 24 +  0 + jn] * mhi;       // 0 | P[d1,0,n-8]
    const float q0n = proj[d1 * 24 +  8 + jn] * mhi;       // 0 | P[d1,1,n-8]
    const float q1n = proj[d1 * 24 + 16 + jn] * mhi;       // 0 | P[d1,2,n-8]

    v2f b0, b1;
    b0[0] = half ? r2n  : r0n;   // vgpr0: row K2 | row K0
    b0[1] = half ? r3n  : r1n;   // vgpr1: row K3 | row K1
    b1[0] = half ? 0.0f : q0n;   // chunk1 rows K2,K3 are zero
    b1[1] = half ? 0.0f : q1n;

    v8f c = {};
    c = __builtin_amdgcn_wmma_f32_16x16x4_f32(false, a0, false, b0,
                                              (short)0, c, false, false);
    c = __builtin_amdgcn_wmma_f32_16x16x4_f32(false, a1, false, b1,
                                              (short)0, c, false, false);

    // D layout: VGPR r -> (M=r, N=lane) for lanes 0-15, (M=r+8, N=lane-16) else.
    // out element index = point*64 + 16*t + N
    float* o = out + (base + (long long)half * 8) * 64 + 16 * t + n;
    if (fullTile) {
      // Hot path: wave-uniform, no per-store exec masking; compiler folds the
      // r*64 element strides into immediate offsets (r*256 bytes).
#pragma unroll
      for (int r = 0; r < 8; ++r)
        __builtin_nontemporal_store(c[r], o + r * 64);
    } else {
      // Tail tile only (never taken when nPoints % 16 == 0).
#pragma unroll
      for (int r = 0; r < 8; ++r) {
        const long long row = base + half * 8 + r;
        if (row < nPoints)
          __builtin_nontemporal_store(c[r], o + r * 64);
      }
    }
  }
#else
  // Fallback (no f32 K=4 WMMA builtin): lower half-lanes write directly.
  if (half == 0 && p < nPoints) {
#pragma unroll
    for (int d = 0; d < SE_DEPTH; ++d)
#pragma unroll
      for (int j = 0; j < 8; ++j)
        __builtin_nontemporal_store(
            rho[d]   * proj[d * 24 +  0 + j] +
            phi[d]   * proj[d * 24 +  8 + j] +
            theta[d] * proj[d * 24 + 16 + j],
            out + p * 64 + d * 8 + j);
  }
#endif
}

extern "C" void kernel_launch(void* const* d_in, const int* in_sizes, int n_in,
                              void* d_out, int out_size, void* d_ws, size_t ws_size,
                              hipStream_t stream) {
  (void)n_in; (void)out_size; (void)d_ws; (void)ws_size;
  const float* x       = (const float*)d_in[0];
  const float* centers = (const float*)d_in[1];
  const float* proj    = (const float*)d_in[2];
  float* out           = (float*)d_out;

  const int nPoints = in_sizes[0] / 3;                 // B = 2,000,000
  const int waves   = (nPoints + 15) / 16;             // 16 points per wave
  const int blocks  = (waves + 7) / 8;                 // 8 waves per 256-thread block

  se_wmma_kernel<<<blocks, 256, 0, stream>>>(x, centers, proj, out, nPoints);
}